// CPANEEKFCPR_16535624089794
// MI455X (gfx1250) — compile-verified
//
#include <hip/hip_runtime.h>
#include <cstdint>
#include <cstddef>

// CPANE-EKF carrier phase recovery: strictly sequential scan over T steps,
// D=4 independent chains. Single wave32, all lanes redundantly carry the 4
// chain states (EXEC must be all-ones for WMMA). Per-step 16-point nearest-
// symbol search done with TWO V_WMMA_F32_16X16X4_F32 (row-rotated A in the
// second) so every lane holds all 16 distances locally -- no ds_bpermute /
// s_wait_dscnt on the serial critical path. Input streamed via async
// global->LDS double buffering (ASYNCcnt).

#define EKF_ALPHA 0.99f
#define EKF_BETA  0.6f
#define CHUNK     128        // steps per LDS chunk (128 * 32B = 4 KB)

typedef __attribute__((ext_vector_type(2))) float v2f;
typedef __attribute__((ext_vector_type(8))) float v8f;

struct C2 { float r, i; };

__device__ __forceinline__ C2 cmul(C2 a, C2 b) {
    return C2{a.r * b.r - a.i * b.i, a.r * b.i + a.i * b.r};
}

struct EKFState {
    C2 psic[4];  // Psi_c posterior phase/amp estimate
    C2 pc[4];    // P_c posterior covariance
    C2 psia[4];  // Psi_a EMA for decision derotation
    C2 q[4];     // Q process noise
    C2 rr[4];    // R measurement noise
};

// Issue 8 x b128 async copies: 32 lanes * 16B * 8 = 4096 B = one chunk.
// GVS mode: addr = SGPR64(base) + VGPR32(byte offset) + imm; the imm offset
// applies to BOTH the global address and the LDS destination address.
__device__ __forceinline__ void issue_async_chunk(const float* sig,
                                                  uint32_t gbyte,
                                                  uint32_t lds_addr_lane) {
    uint32_t voff = gbyte + (uint32_t)threadIdx.x * 16u;
    asm volatile(
        "global_load_async_to_lds_b128 %0, %1, %2 offset:0\n\t"
        "global_load_async_to_lds_b128 %0, %1, %2 offset:512\n\t"
        "global_load_async_to_lds_b128 %0, %1, %2 offset:1024\n\t"
        "global_load_async_to_lds_b128 %0, %1, %2 offset:1536\n\t"
        "global_load_async_to_lds_b128 %0, %1, %2 offset:2048\n\t"
        "global_load_async_to_lds_b128 %0, %1, %2 offset:2560\n\t"
        "global_load_async_to_lds_b128 %0, %1, %2 offset:3072\n\t"
        "global_load_async_to_lds_b128 %0, %1, %2 offset:3584\n\t"
        :: "v"(lds_addr_lane), "v"(voff), "s"(sig)
        : "memory");
}

// One EKF time step for all 4 dims (executed redundantly by all 32 lanes so
// EXEC is all-ones at the WMMAs). aMat / aMat2 are the constant constellation
// A matrices (aMat2 rows rotated by 8 so each lane sees all 16 rows).
__device__ __forceinline__ void ekf_step(EKFState& S, const C2 y[4],
                                         v2f aMat, v2f aMat2, int lane,
                                         float2* __restrict__ outp) {
    const float S10 = 0.31622776601683794f;  // 1/sqrt(10): 16QAM unit-power scale

    float yrr[4], yri[4], cosp[4], sinp[4], emi[4];
#pragma unroll
    for (int d = 0; d < 4; ++d) {
        // Psi_a EMA (uses prior Psi_c)
        S.psia[d].r = EKF_BETA * S.psia[d].r + (1.0f - EKF_BETA) * S.psic[d].r;
        S.psia[d].i = EKF_BETA * S.psia[d].i + (1.0f - EKF_BETA) * S.psic[d].i;
        // y_rot = y * exp(-i*Psi_a), Psi_a complex: exp(-i(a+ib)) = e^b (cos a - i sin a)
        float er = __expf(S.psia[d].i);
        float sa, ca;
        __sincosf(S.psia[d].r, &sa, &ca);
        C2 rot{er * ca, -er * sa};
        C2 yr = cmul(y[d], rot);
        yrr[d] = yr.r;
        yri[d] = yr.i;
        // e = exp(+i*Psi_p), Psi_p = prior Psi_c: e^{-Im} (cos Re + i sin Re)
        emi[d] = __expf(-S.psic[d].i);
        __sincosf(S.psic[d].r, &sinp[d], &cosp[d]);
    }

    // ---- nearest constellation point via 2x V_WMMA_F32_16X16X4_F32 ----
    // D[m][n] = |c_m|^2 - 2 Re(c_m) yr_n - 2 Im(c_m) yi_n  (== dist^2 - |y|^2)
    // B layout (4x16, K x N): lanes 0-15 hold {K=0,K=1}, lanes 16-31 {K=2,K=3}
    int nl = lane & 15;
    bool low = lane < 16;
    bool valid = nl < 4;
    float selr = (nl == 0) ? yrr[0] : (nl == 1) ? yrr[1] : (nl == 2) ? yrr[2] : yrr[3];
    float seli = (nl == 0) ? yri[0] : (nl == 1) ? yri[1] : (nl == 2) ? yri[2] : yri[3];
    v2f b;
    b.x = low ? (valid ? 1.0f : 0.0f) : (valid ? -2.0f * seli : 0.0f);
    b.y = low ? (valid ? -2.0f * selr : 0.0f) : 0.0f;
    v8f cz = {0.f, 0.f, 0.f, 0.f, 0.f, 0.f, 0.f, 0.f};
    // WMMA1: rows c_0..c_15 -> low lanes get rows 0-7, high lanes rows 8-15
    v8f d1 = __builtin_amdgcn_wmma_f32_16x16x4_f32(
        false, aMat, false, b, (short)0, cz, false, false);
    // WMMA2: rows c_8..c_15,c_0..c_7 -> low lanes get rows 8-15, high rows 0-7
    v8f d2 = __builtin_amdgcn_wmma_f32_16x16x4_f32(
        false, aMat2, false, b, (short)0, cz, false, false);

    // per-lane 16-way argmin, tree reduction (depth 4), ties keep the lower slot of each pair.
    // Low-lane label convention: d1[r] -> row r, d2[r] -> row r+8.
    // High lanes hold the same 16 values with labels xor 8 (fixed at the end).
    float val[16];
    int lab[16];
#pragma unroll
    for (int k = 0; k < 8; ++k) {
        val[k] = d1[k];     lab[k] = k;
        val[k + 8] = d2[k]; lab[k + 8] = k + 8;
    }
#pragma unroll
    for (int k = 0; k < 8; ++k)
        if (val[k + 8] < val[k]) { val[k] = val[k + 8]; lab[k] = lab[k + 8]; }
#pragma unroll
    for (int k = 0; k < 4; ++k)
        if (val[k + 4] < val[k]) { val[k] = val[k + 4]; lab[k] = lab[k + 4]; }
#pragma unroll
    for (int k = 0; k < 2; ++k)
        if (val[k + 2] < val[k]) { val[k] = val[k + 2]; lab[k] = lab[k + 2]; }
    if (val[1] < val[0]) { val[0] = val[1]; lab[0] = lab[1]; }
    int row = low ? lab[0] : (lab[0] ^ 8);

    int idxs[4];
    idxs[0] = __builtin_amdgcn_readlane(row, 0);
    idxs[1] = __builtin_amdgcn_readlane(row, 1);
    idxs[2] = __builtin_amdgcn_readlane(row, 2);
    idxs[3] = __builtin_amdgcn_readlane(row, 3);

    C2 outv[4];
#pragma unroll
    for (int d = 0; d < 4; ++d) {
        int idx = idxs[d];
        // decode 16QAM point: c = ((idx>>2)*2-3, (idx&3)*2-3)/sqrt(10)
        C2 dd{(float)(2 * (idx >> 2) - 3) * S10, (float)(2 * (idx & 3) - 3) * S10};
        C2 e{emi[d] * cosp[d], emi[d] * sinp[d]};   // exp(i*Psi_p)
        C2 de = cmul(dd, e);
        C2 H{-de.i, de.r};                          // i * d * e
        C2 Pp{S.pc[d].r + S.q[d].r, S.pc[d].i + S.q[d].i};
        float h2 = H.r * H.r + H.i * H.i;           // H*conj(H) is real
        C2 HPH{h2 * Pp.r, h2 * Pp.i};
        C2 den{HPH.r + S.rr[d].r, HPH.i + S.rr[d].i};
        float inv = __builtin_amdgcn_rcpf(den.r * den.r + den.i * den.i);
        C2 num = cmul(Pp, C2{H.r, -H.i});           // P_p * conj(H)
        C2 K{(num.r * den.r + num.i * den.i) * inv,
             (num.i * den.r - num.r * den.i) * inv};
        C2 v{y[d].r - de.r, y[d].i - de.i};         // innovation
        C2 Kv = cmul(K, v);

        // output: phi = prior Psi_c = Psi_p -> reuse sincos; exp(+Im) = rcp(exp(-Im))
        float ep = __builtin_amdgcn_rcpf(emi[d]);
        C2 ro{ep * cosp[d], -ep * sinp[d]};         // exp(-i*phi)
        outv[d] = cmul(y[d], ro);

        // state updates
        C2 KH = cmul(K, H);
        C2 om{1.0f - KH.r, -KH.i};
        S.pc[d] = cmul(om, Pp);
        float kv2 = Kv.r * Kv.r + Kv.i * Kv.i;      // Kv*conj(Kv) is real
        S.q[d].r = EKF_ALPHA * S.q[d].r + (1.0f - EKF_ALPHA) * kv2;
        S.q[d].i = EKF_ALPHA * S.q[d].i;
        float vv = v.r * v.r + v.i * v.i;           // v*conj(v) is real
        S.rr[d].r = EKF_ALPHA * S.rr[d].r + (1.0f - EKF_ALPHA) * (vv - HPH.r);
        S.rr[d].i = EKF_ALPHA * S.rr[d].i + (1.0f - EKF_ALPHA) * (-HPH.i);
        S.psic[d].r += Kv.r;
        S.psic[d].i += Kv.i;
    }

    // lanes 0..3 store their dim's output (complex64 as float2)
    float orr = (lane == 0) ? outv[0].r : (lane == 1) ? outv[1].r
              : (lane == 2) ? outv[2].r : outv[3].r;
    float ori = (lane == 0) ? outv[0].i : (lane == 1) ? outv[1].i
              : (lane == 2) ? outv[2].i : outv[3].i;
    if (lane < 4) outp[lane] = make_float2(orr, ori);
}

__global__ __launch_bounds__(32) void cpane_ekf_kernel(
    const float* __restrict__ sig, float2* __restrict__ outp, int T) {
    __shared__ __align__(16) float4 sbuf[2][CHUNK * 2];
    const int lane = threadIdx.x;
    const float S10 = 0.31622776601683794f;

    // Constant A matrices (16x4, M x K): A[m] = {|c_m|^2, Re c_m, Im c_m, 0}
    // Layout: lanes 0-15 hold {K=0,K=1}, lanes 16-31 hold {K=2,K=3}, M = lane&15.
    // aMat2 uses rows rotated by 8: A2[m] = A[(m+8) & 15].
    int m = lane & 15;
    int m2 = (m + 8) & 15;
    float cre  = (float)(2 * (m  >> 2) - 3) * S10;
    float cim  = (float)(2 * (m  &  3) - 3) * S10;
    float cre2 = (float)(2 * (m2 >> 2) - 3) * S10;
    float cim2 = (float)(2 * (m2 &  3) - 3) * S10;
    v2f aMat, aMat2;
    if (lane < 16) {
        aMat.x  = cre  * cre  + cim  * cim;  aMat.y  = cre;
        aMat2.x = cre2 * cre2 + cim2 * cim2; aMat2.y = cre2;
    } else {
        aMat.x  = cim;  aMat.y  = 0.0f;
        aMat2.x = cim2; aMat2.y = 0.0f;
    }

    EKFState S;
#pragma unroll
    for (int d = 0; d < 4; ++d) {
        S.psic[d] = C2{0.0f, 0.0f};
        S.pc[d]   = C2{1e-5f, 1e-5f};
        S.psia[d] = C2{0.0f, 0.0f};
        S.q[d]    = C2{1e-5f, 1e-5f};
        S.rr[d]   = C2{1e-2f, 1e-2f};
    }

    uint32_t lds0 = (uint32_t)(uintptr_t)&sbuf[0][0] + (uint32_t)lane * 16u;
    uint32_t lds1 = (uint32_t)(uintptr_t)&sbuf[1][0] + (uint32_t)lane * 16u;

    int full = T / CHUNK;
    if (full > 0) issue_async_chunk(sig, 0u, lds0);

    for (int c = 0; c < full; ++c) {
        if (c + 1 < full) {
            issue_async_chunk(sig, (uint32_t)(c + 1) * (CHUNK * 32u),
                              ((c + 1) & 1) ? lds1 : lds0);
            // two chunks in flight (16 ops); <=8 outstanding => chunk c done
            asm volatile("s_wait_asynccnt 8" ::: "memory");
        } else {
            asm volatile("s_wait_asynccnt 0" ::: "memory");
        }
        const float4* buf = sbuf[c & 1];
        for (int i = 0; i < CHUNK; ++i) {
            float4 p0 = buf[2 * i];
            float4 p1 = buf[2 * i + 1];
            C2 y[4] = {{p0.x, p0.y}, {p0.z, p0.w}, {p1.x, p1.y}, {p1.z, p1.w}};
            ekf_step(S, y, aMat, aMat2, lane,
                     outp + (size_t)(c * CHUNK + i) * 4);
        }
    }
    // tail (T not multiple of CHUNK): direct global loads
    for (int t = full * CHUNK; t < T; ++t) {
        const float4* g = (const float4*)sig + 2 * (size_t)t;
        float4 p0 = g[0];
        float4 p1 = g[1];
        C2 y[4] = {{p0.x, p0.y}, {p0.z, p0.w}, {p1.x, p1.y}, {p1.z, p1.w}};
        ekf_step(S, y, aMat, aMat2, lane, outp + (size_t)t * 4);
    }
}

extern "C" void kernel_launch(void* const* d_in, const int* in_sizes, int n_in,
                              void* d_out, int out_size, void* d_ws, size_t ws_size,
                              hipStream_t stream) {
    (void)n_in; (void)out_size; (void)d_ws; (void)ws_size;
    const float* sig = (const float*)d_in[0];  // complex64 [T,4] as interleaved floats
    float2* outp = (float2*)d_out;             // complex64 [T,4] out
    int T = in_sizes[0] / 4;                   // flat complex element count = T*D
    cpane_ekf_kernel<<<1, 32, 0, stream>>>(sig, outp, T);
}